// MovingNCA_58420145160544
// MI455X (gfx1250) — compile-verified
//
#include <hip/hip_runtime.h>

typedef float v2f __attribute__((ext_vector_type(2)));
typedef float v8f __attribute__((ext_vector_type(8)));

#define NSIDE   512
#define NEO     510
#define CELLS   (NEO * NEO)        // 260100
#define ST_CH   26
#define ODIM    28
#define FEATN   245
#define KP      256                // padded K / LDS row stride
#define NCA_THRESH  0.0007f

// ---------------------------------------------------------------------------
// prep: zero-padded, k-pair-interleaved weight copies.
//   W1q[(k>>1)*512 + n*2 + (k&1)]  (256x256 logical)  -> b64-loadable B frags
//   W2q[(k>>1)*64  + n*2 + (k&1)]  (256x32  logical)
//   b1p[256], b2p[32]
// ---------------------------------------------------------------------------
__global__ void nca_prep(const float* __restrict__ W1, const float* __restrict__ b1,
                         const float* __restrict__ W2, const float* __restrict__ b2,
                         float* __restrict__ W1q, float* __restrict__ W2q,
                         float* __restrict__ b1p, float* __restrict__ b2p) {
    int idx = blockIdx.x * 256 + threadIdx.x;          // 0 .. 65535
    if (idx < 256 * 256) {
        int k = ((idx >> 9) << 1) | (idx & 1);
        int n = (idx >> 1) & 255;
        W1q[idx] = (k < FEATN && n < FEATN) ? W1[k * FEATN + n] : 0.0f;
    }
    if (idx < 256 * 32) {
        int k = ((idx >> 6) << 1) | (idx & 1);
        int n = (idx >> 1) & 31;
        W2q[idx] = (k < FEATN && n < ODIM) ? W2[k * ODIM + n] : 0.0f;
    }
    if (idx < 256) b1p[idx] = (idx < FEATN) ? b1[idx] : 0.0f;
    if (idx < 32)  b2p[idx] = (idx < ODIM)  ? b2[idx] : 0.0f;
}

// ---------------------------------------------------------------------------
// Fused per-iteration kernel: feature gather -> GEMM1 (245x245) -> GEMM2
// (245x28), all through V_WMMA_F32_16X16X4_F32. 4 waves / block, 64 cells.
// GEMM1 split: wave w owns ALL 64 rows x n-tiles [4w..4w+3]; B fragments are
// single b64 loads thanks to the interleaved weight layout.
// ---------------------------------------------------------------------------
__global__ __launch_bounds__(128)
void nca_gemm(const float* __restrict__ img, const float* __restrict__ state,
              const int* __restrict__ perc, const float* __restrict__ W1q,
              const float* __restrict__ b1p, const float* __restrict__ W2q,
              const float* __restrict__ b2p, float* __restrict__ guesses) {
    __shared__ float tile[64 * KP];                    // 64 KB: features, then hidden

    const int tid   = threadIdx.x;
    const int cell0 = blockIdx.x * 64;

    // ---- Phase A: build 64 feature rows (uniform trip count: EXEC stays full)
    #pragma unroll 1
    for (int t = 0; t < (64 * KP) / 128; ++t) {
        int idx  = t * 128 + tid;
        int row  = idx >> 8;
        int f    = idx & 255;
        int cell = cell0 + row;
        if (cell >= CELLS) cell = CELLS - 1;           // clamp (stores predicated later)
        int ci = cell / NEO;
        int cj = cell - ci * NEO;
        float v = 0.0f;
        if (f < 243) {
            int pos = f / 27;                          // 3x3 position 0..8
            int c   = f - pos * 27;                    // channel 0..26
            int di  = pos / 3, dj = pos - di * 3;
            if (c == 0) {
                int px = perc[cell * 2 + 0] + di;
                int py = perc[cell * 2 + 1] + dj;
                v = img[px * NSIDE + py];
            } else {
                v = state[((ci + di) * NSIDE + (cj + dj)) * ST_CH + (c - 1)];
            }
        } else if (f == 243) {
            v = (float)(perc[cell * 2 + 0] - (NSIDE / 2)) * (1.0f / (NSIDE / 2));
        } else if (f == 244) {
            v = (float)(perc[cell * 2 + 1] - (NSIDE / 2)) * (1.0f / (NSIDE / 2));
        }                                              // f in 245..255 -> 0 (K padding)
        tile[row * KP + f] = v;
    }
    __syncthreads();

    const int lane = tid & 31;
    const int wave = tid >> 5;
    const int ln   = lane & 15;                        // N (or M) index within tile
    const int hs   = lane >> 4;                        // K half-select
    const int nb0  = wave * 64;                        // first output column of this wave

    // ---- GEMM1: full 64 rows x this wave's 64 columns
    v8f acc[4][4];                                     // [row-tile][n-tile]
    #pragma unroll
    for (int rt = 0; rt < 4; ++rt)
        #pragma unroll
        for (int nt = 0; nt < 4; ++nt)
            acc[rt][nt] = (v8f){0.f,0.f,0.f,0.f,0.f,0.f,0.f,0.f};

    #pragma unroll 2
    for (int kt = 0; kt < 62; ++kt) {                  // 62*4 = 248 >= 245 (zero-padded)
        int k  = kt * 4 + hs * 2;                      // A k-index
        int k2 = kt * 2 + hs;                          // interleaved B row-pair index
        const float* bp = W1q + (size_t)k2 * 512 + (nb0 + ln) * 2;
        v2f b[4];
        #pragma unroll
        for (int nt = 0; nt < 4; ++nt)
            b[nt] = *(const v2f*)(bp + nt * 32);       // one b64 per n-tile
        #pragma unroll
        for (int rt = 0; rt < 4; ++rt) {
            v2f a;
            a.x = tile[(rt * 16 + ln) * KP + k];
            a.y = tile[(rt * 16 + ln) * KP + k + 1];
            #pragma unroll
            for (int nt = 0; nt < 4; ++nt)
                acc[rt][nt] = __builtin_amdgcn_wmma_f32_16x16x4_f32(
                    false, a, false, b[nt], (short)0, acc[rt][nt], false, false);
        }
    }
    // bias b1 (broadcast over rows; column = nb0 + nt*16 + ln)
    #pragma unroll
    for (int nt = 0; nt < 4; ++nt) {
        float bb = b1p[nb0 + nt * 16 + ln];
        #pragma unroll
        for (int rt = 0; rt < 4; ++rt)
            #pragma unroll
            for (int r = 0; r < 8; ++r) acc[rt][nt][r] += bb;
    }

    __syncthreads();                                   // all waves done reading features
    // write hidden to LDS (C layout: VGPR r -> M=r / M=r+8)
    #pragma unroll
    for (int rt = 0; rt < 4; ++rt)
        #pragma unroll
        for (int nt = 0; nt < 4; ++nt)
            #pragma unroll
            for (int r = 0; r < 8; ++r) {
                int m = rt * 16 + r + hs * 8;
                tile[m * KP + nb0 + nt * 16 + ln] = acc[rt][nt][r];
            }
    __syncthreads();

    // ---- GEMM2: wave w -> rows [16w..16w+15] x 28 cols, K = 256 (pad cols are 0)
    const int arow = wave * 16 + ln;
    v8f acc2[2];
    acc2[0] = (v8f){0.f,0.f,0.f,0.f,0.f,0.f,0.f,0.f};
    acc2[1] = acc2[0];
    #pragma unroll 2
    for (int kt = 0; kt < 64; ++kt) {
        int k  = kt * 4 + hs * 2;
        int k2 = kt * 2 + hs;
        v2f a;
        a.x = tile[arow * KP + k];
        a.y = tile[arow * KP + k + 1];
        #pragma unroll
        for (int nt = 0; nt < 2; ++nt) {
            v2f b = *(const v2f*)(W2q + (size_t)k2 * 64 + (nt * 16 + ln) * 2);
            acc2[nt] = __builtin_amdgcn_wmma_f32_16x16x4_f32(
                false, a, false, b, (short)0, acc2[nt], false, false);
        }
    }
    // bias b2 + predicated store of guesses
    #pragma unroll
    for (int nt = 0; nt < 2; ++nt) {
        float bb = b2p[nt * 16 + ln];
        #pragma unroll
        for (int r = 0; r < 8; ++r) {
            int m    = r + hs * 8;
            int cell = cell0 + wave * 16 + m;
            int n2   = nt * 16 + ln;
            if (cell < CELLS && n2 < ODIM)
                guesses[cell * ODIM + n2] = acc2[nt][r] + bb;
        }
    }
}

// ---------------------------------------------------------------------------
// State accumulate (non-overlapping writes) + perception threshold update.
// ---------------------------------------------------------------------------
__global__ void nca_update(float* __restrict__ state, int* __restrict__ perc,
                           const float* __restrict__ guesses) {
    int cell = blockIdx.x * 256 + threadIdx.x;
    if (cell >= CELLS) return;
    int i = cell / NEO, j = cell - i * NEO;
    float* s = state + ((i + 1) * NSIDE + (j + 1)) * ST_CH;
    const float* g = guesses + cell * ODIM;
    #pragma unroll
    for (int c = 0; c < ST_CH; ++c) s[c] += g[c];

    float ax = g[ST_CH], ay = g[ST_CH + 1];
    int dx = (ax < -NCA_THRESH) ? -1 : ((ax > NCA_THRESH) ? 1 : 0);
    int dy = (ay < -NCA_THRESH) ? -1 : ((ay > NCA_THRESH) ? 1 : 0);
    int px = perc[cell * 2 + 0] + dx;
    int py = perc[cell * 2 + 1] + dy;
    px = px < 0 ? 0 : (px > NEO - 1 ? NEO - 1 : px);
    py = py < 0 ? 0 : (py > NEO - 1 ? NEO - 1 : py);
    perc[cell * 2 + 0] = px;
    perc[cell * 2 + 1] = py;
}

// ---------------------------------------------------------------------------
// Output #1: state[1:511, 1:511, 16:26]  (510*510*10 floats)
// ---------------------------------------------------------------------------
__global__ void nca_finalize(const float* __restrict__ state, float* __restrict__ out) {
    int idx = blockIdx.x * 256 + threadIdx.x;
    if (idx >= CELLS * 10) return;
    int cell = idx / 10, c = idx - cell * 10;
    int i = cell / NEO, j = cell - i * NEO;
    out[idx] = state[((i + 1) * NSIDE + (j + 1)) * ST_CH + 16 + c];
}

// ---------------------------------------------------------------------------
extern "C" void kernel_launch(void* const* d_in, const int* in_sizes, int n_in,
                              void* d_out, int out_size, void* d_ws, size_t ws_size,
                              hipStream_t stream) {
    const float* img    = (const float*)d_in[0];   // 512*512*1
    const float* state0 = (const float*)d_in[1];   // 512*512*26
    const int*   perc0  = (const int*)  d_in[2];   // 510*510*2
    const float* W1     = (const float*)d_in[3];   // 245*245
    const float* b1     = (const float*)d_in[4];   // 245
    const float* W2     = (const float*)d_in[5];   // 245*28
    const float* b2     = (const float*)d_in[6];   // 28

    auto align256 = [](size_t x) { return (x + 255) & ~(size_t)255; };
    char*  ws  = (char*)d_ws;
    size_t off = 0;
    float* state = (float*)(ws + off); off = align256(off + (size_t)NSIDE * NSIDE * ST_CH * 4);
    int*   perc  = (int*)  (ws + off); off = align256(off + (size_t)CELLS * 2 * 4);
    float* gbuf  = (float*)(ws + off); off = align256(off + (size_t)CELLS * ODIM * 4);
    float* W1q   = (float*)(ws + off); off = align256(off + (size_t)256 * 256 * 4);
    float* W2q   = (float*)(ws + off); off = align256(off + (size_t)256 * 32 * 4);
    float* b1p   = (float*)(ws + off); off = align256(off + (size_t)256 * 4);
    float* b2p   = (float*)(ws + off); off = align256(off + (size_t)32 * 4);
    (void)off; (void)ws_size; (void)in_sizes; (void)n_in; (void)out_size;

    // live state/perception copies (inputs must not be mutated)
    hipMemcpyAsync(state, state0, (size_t)NSIDE * NSIDE * ST_CH * 4,
                   hipMemcpyDeviceToDevice, stream);
    hipMemcpyAsync(perc, perc0, (size_t)CELLS * 2 * 4,
                   hipMemcpyDeviceToDevice, stream);

    nca_prep<<<256, 256, 0, stream>>>(W1, b1, W2, b2, W1q, W2q, b1p, b2p);

    const int gemm_blocks = (CELLS + 63) / 64;     // 4065
    for (int it = 0; it < 4; ++it) {
        nca_gemm<<<gemm_blocks, 128, 0, stream>>>(img, state, perc,
                                                  W1q, b1p, W2q, b2p, gbuf);
        nca_update<<<(CELLS + 255) / 256, 256, 0, stream>>>(state, perc, gbuf);
    }

    nca_finalize<<<(CELLS * 10 + 255) / 256, 256, 0, stream>>>(state, (float*)d_out);
    hipMemcpyAsync((float*)d_out + (size_t)CELLS * 10, gbuf,
                   (size_t)CELLS * ODIM * 4, hipMemcpyDeviceToDevice, stream);
}